// Self_Attn_1082331758709
// MI455X (gfx1250) — compile-verified
//
#include <hip/hip_runtime.h>
#include <math.h>
#include <stdint.h>

typedef float v2f __attribute__((ext_vector_type(2)));
typedef float v8f __attribute__((ext_vector_type(8)));

#define BN   4
#define CH   64
#define NN   4096
#define DQK  8
#define JT   4              // 16-col sub-tiles per flash chunk (64 cols)
#define JCHUNK (JT * 16)
#define WAVES  4            // i-tiles (waves) per workgroup

// ---------------------------------------------------------------------------
// Stage 1: 1x1 convs.  qT/kT[B][N][8]; V in pair-interleaved layout
//   vP[b][j>>1][c][j&1]  so rows (j, j+1) at same c form one b64.
// ---------------------------------------------------------------------------
__global__ __launch_bounds__(256) void qkv_kernel(
    const float* __restrict__ x,
    const float* __restrict__ wq, const float* __restrict__ bq,
    const float* __restrict__ wk, const float* __restrict__ bk,
    const float* __restrict__ wv, const float* __restrict__ bv,
    float* __restrict__ qT, float* __restrict__ kT, float* __restrict__ vP)
{
    __shared__ float swq[DQK * CH];
    __shared__ float swk[DQK * CH];
    __shared__ float swv[CH * CH];
    __shared__ float sbq[DQK], sbk[DQK], sbv[CH];

    const int tid = threadIdx.x;
    for (int i = tid; i < CH * CH; i += 256) swv[i] = wv[i];
    for (int i = tid; i < DQK * CH; i += 256) { swq[i] = wq[i]; swk[i] = wk[i]; }
    if (tid < DQK) { sbq[tid] = bq[tid]; sbk[tid] = bk[tid]; }
    if (tid < CH)  { sbv[tid] = bv[tid]; }
    __syncthreads();

    const int b = blockIdx.x >> 4;                 // NN/256 = 16 chunks/batch
    const int n = ((blockIdx.x & 15) << 8) + tid;  // pixel index

    float qa[DQK] = {0.f}, ka[DQK] = {0.f};
    float va[CH];
#pragma unroll
    for (int o = 0; o < CH; ++o) va[o] = 0.f;

    const float* xp = x + (size_t)b * CH * NN + n;
#pragma unroll 4
    for (int c = 0; c < CH; ++c) {
        const float xv = xp[(size_t)c * NN];
#pragma unroll
        for (int d = 0; d < DQK; ++d) {
            qa[d] += swq[d * CH + c] * xv;
            ka[d] += swk[d * CH + c] * xv;
        }
#pragma unroll
        for (int o = 0; o < CH; ++o) va[o] += swv[o * CH + c] * xv;
    }

    const size_t base8 = ((size_t)b * NN + n) * DQK;
#pragma unroll
    for (int d = 0; d < DQK; ++d) {
        qT[base8 + d] = qa[d] + sbq[d];
        kT[base8 + d] = ka[d] + sbk[d];
    }
    const size_t vbase = (((size_t)b * (NN / 2) + (n >> 1)) * CH) * 2 + (n & 1);
#pragma unroll
    for (int o = 0; o < CH; ++o) vP[vbase + 2 * o] = va[o] + sbv[o];
}

// ---------------------------------------------------------------------------
// Stage 2: flash attention. 4 waves/block, one 16-row i-tile per wave,
// all waves share each 64-col V chunk staged in LDS via async b128 copy.
// Softmax split: max-cascade for chunk, then per-tile exp->bounce->PV wmma
// (keeps e[t] liveness short -> no VGPR-MSB pressure), sum-cascade at end.
// ---------------------------------------------------------------------------
__global__ __launch_bounds__(WAVES * 32) void flash_attn_kernel(
    const float* __restrict__ qT, const float* __restrict__ kT,
    const float* __restrict__ vP, const float* __restrict__ x,
    const float* __restrict__ gamma, float* __restrict__ out)
{
    __shared__ float Vlds[JCHUNK * CH];        // 16 KB shared V chunk
    __shared__ float Plds[WAVES * 16 * 16];    // 1 KB per-wave P bounce

    const int tid  = threadIdx.x;
    const int wv   = tid >> 5;
    const int lane = tid & 31;
    const int lp   = lane & 15;
    const int half = lane >> 4;

    const int b  = blockIdx.x >> 6;                   // 64 blocks per batch
    const int it = ((blockIdx.x & 63) << 2) + wv;     // i-tile of this wave

    // A operands for Q: lane holds row M=lp, Ks = half*2+{0,1} (+4 for 2nd op)
    const float* qrow = qT + ((size_t)b * NN + it * 16 + lp) * DQK + half * 2;
    const v2f aq0 = *(const v2f*)(qrow);
    const v2f aq1 = *(const v2f*)(qrow + 4);

    v8f O[4];
#pragma unroll
    for (int cb = 0; cb < 4; ++cb) O[cb] = (v8f){0.f,0.f,0.f,0.f,0.f,0.f,0.f,0.f};
    float m[8], l[8];
#pragma unroll
    for (int r = 0; r < 8; ++r) { m[r] = -__builtin_inff(); l[r] = 0.f; }

    const float* kbase = kT + (size_t)b * NN * DQK;
    const float* vbase = vP + (size_t)b * NN * CH;    // pair-interleaved

    // byte bases for async copy (this wave's 4KB quarter of the 16KB chunk)
    const uint32_t vlds_base = (uint32_t)(size_t)(void*)&Vlds[0]
                             + (uint32_t)(wv * 4096 + lane * 16);
    float* pbounce = &Plds[wv * 256];

    for (int jc = 0; jc < NN / JCHUNK; ++jc) {
        const int jbase = jc * JCHUNK;

        __syncthreads();   // everyone done reading previous V chunk

        // ---- async copy this chunk's V (16KB) into LDS ------------------
        {
            const uint64_t gsrc = (uint64_t)(size_t)(const void*)
                                  (vbase + (size_t)jbase * CH)
                                + (uint64_t)(wv * 4096 + lane * 16);
#pragma unroll
            for (int i = 0; i < 8; ++i) {
                const uint32_t la = vlds_base + (uint32_t)(i * 512);
                const uint64_t ga = gsrc + (uint64_t)(i * 512);
                asm volatile("global_load_async_to_lds_b128 %0, %1, off"
                             :: "v"(la), "v"(ga) : "memory");
            }
        }

        // ---- energy chunk: E[t] = Q x K_t  (8 wmma, overlaps copy) ------
        v8f e[JT];
#pragma unroll
        for (int t = 0; t < JT; ++t) {
            const int jcol = jbase + t * 16 + lp;
            const float* krow = kbase + (size_t)jcol * DQK + half * 2;
            const v2f bk0 = *(const v2f*)(krow);
            const v2f bk1 = *(const v2f*)(krow + 4);
            v8f z = (v8f){0.f,0.f,0.f,0.f,0.f,0.f,0.f,0.f};
            z = __builtin_amdgcn_wmma_f32_16x16x4_f32(false, aq0, false, bk0,
                                                      (short)0, z, false, false);
            e[t] = __builtin_amdgcn_wmma_f32_16x16x4_f32(false, aq1, false, bk1,
                                                         (short)0, z, false, false);
        }

        // ---- phase 1: row max cascade + rescale accumulators ------------
        float alpha[8], rsacc[8];
#pragma unroll
        for (int r = 0; r < 8; ++r) {
            float tm = e[0][r];
#pragma unroll
            for (int t = 1; t < JT; ++t) tm = fmaxf(tm, e[t][r]);
            tm = fmaxf(tm, __shfl_xor(tm, 1, 32));
            tm = fmaxf(tm, __shfl_xor(tm, 2, 32));
            tm = fmaxf(tm, __shfl_xor(tm, 4, 32));
            tm = fmaxf(tm, __shfl_xor(tm, 8, 32));
            const float mn = fmaxf(m[r], tm);
            alpha[r] = __expf(m[r] - mn);     // 0 on first chunk (m=-inf)
            m[r] = mn;
            rsacc[r] = 0.f;
        }
#pragma unroll
        for (int cb = 0; cb < 4; ++cb)
#pragma unroll
            for (int r = 0; r < 8; ++r) O[cb][r] *= alpha[r];

        // ---- V chunk must be resident before PV -------------------------
        asm volatile("s_wait_asynccnt 0x0" ::: "memory");
        __syncthreads();

        // ---- phase 2: per tile  exp -> LDS bounce -> 16 PV wmma ---------
#pragma unroll
        for (int t = 0; t < JT; ++t) {
#pragma unroll
            for (int r = 0; r < 8; ++r) {
                const float p = __expf(e[t][r] - m[r]);
                rsacc[r] += p;
                pbounce[(r + half * 8) * 16 + lp] = p;   // e[t] dies here
            }
            v2f pa[4];
#pragma unroll
            for (int kb = 0; kb < 4; ++kb)
                pa[kb] = *(const v2f*)&pbounce[lp * 16 + kb * 4 + half * 2];
#pragma unroll
            for (int cb = 0; cb < 4; ++cb) {
                const int c = cb * 16 + lp;
#pragma unroll
                for (int kb = 0; kb < 4; ++kb) {
                    const int lr = t * 16 + kb * 4 + half * 2;  // even
                    const v2f vb = *(const v2f*)&Vlds[((lr >> 1) * CH + c) * 2];
                    O[cb] = __builtin_amdgcn_wmma_f32_16x16x4_f32(
                                false, pa[kb], false, vb, (short)0, O[cb],
                                false, false);
                }
            }
        }

        // ---- phase 3: finish row sums, update l -------------------------
#pragma unroll
        for (int r = 0; r < 8; ++r) {
            float rs = rsacc[r];
            rs += __shfl_xor(rs, 1, 32);
            rs += __shfl_xor(rs, 2, 32);
            rs += __shfl_xor(rs, 4, 32);
            rs += __shfl_xor(rs, 8, 32);
            l[r] = l[r] * alpha[r] + rs;
        }
    }

    // ---- epilogue: out = gamma * O/l + x --------------------------------
    const float g = gamma[0];
#pragma unroll
    for (int r = 0; r < 8; ++r) {
        const int i = it * 16 + r + half * 8;
        const float inv = 1.0f / l[r];
#pragma unroll
        for (int cb = 0; cb < 4; ++cb) {
            const int c = cb * 16 + lp;
            const size_t idx = ((size_t)b * CH + c) * NN + i;
            out[idx] = g * (O[cb][r] * inv) + x[idx];
        }
    }
}

// ---------------------------------------------------------------------------
extern "C" void kernel_launch(void* const* d_in, const int* in_sizes, int n_in,
                              void* d_out, int out_size, void* d_ws, size_t ws_size,
                              hipStream_t stream)
{
    (void)in_sizes; (void)n_in; (void)out_size; (void)ws_size;
    const float* x     = (const float*)d_in[0];
    const float* wq    = (const float*)d_in[1];
    const float* bq    = (const float*)d_in[2];
    const float* wk    = (const float*)d_in[3];
    const float* bk    = (const float*)d_in[4];
    const float* wv    = (const float*)d_in[5];
    const float* bv    = (const float*)d_in[6];
    const float* gamma = (const float*)d_in[7];
    float* out = (float*)d_out;

    float* qT = (float*)d_ws;                          // [B][N][8]
    float* kT = qT + (size_t)BN * NN * DQK;            // [B][N][8]
    float* vP = kT + (size_t)BN * NN * DQK;            // [B][N/2][64][2]

    qkv_kernel<<<BN * (NN / 256), 256, 0, stream>>>(
        x, wq, bq, wk, bk, wv, bv, qT, kT, vP);

    flash_attn_kernel<<<(BN * (NN / 16)) / WAVES, WAVES * 32, 0, stream>>>(
        qT, kT, vP, x, gamma, out);
}